// CyclicShiftConv_58265526337591
// MI455X (gfx1250) — compile-verified
//
#include <hip/hip_runtime.h>

// Problem constants (from reference)
constexpr int Bn   = 16;
constexpr int Cn   = 256;
constexpr int Nn   = 64;
constexpr int Ln   = Nn * Nn;   // 4096
constexpr int REDn = 16;
constexpr int Rn   = 4;
constexpr int Mrows = Bn * Rn;  // 64 rows for the MLP GEMMs (b major, r minor)

typedef __attribute__((ext_vector_type(2))) float v2f;
typedef __attribute__((ext_vector_type(8))) float v8f;

typedef unsigned int u32;
typedef __attribute__((ext_vector_type(4))) unsigned int u32x4;
typedef __attribute__((ext_vector_type(4))) int          i32x4;
typedef __attribute__((ext_vector_type(8))) int          i32x8;

// ---------------------------------------------------------------------------
// Kernel 1: per-(b,c) mean of x over L.  rot_idx rows are permutations, so the
// gathered mean equals the plain mean -> compute once, not once per rotation.
// One 256-thread block per row (4096 rows), float4 streaming loads.
// ---------------------------------------------------------------------------
__global__ __launch_bounds__(256) void mean_kernel(const float* __restrict__ x,
                                                   float* __restrict__ s_mean) {
    const int row = blockIdx.x;                       // b*Cn + c
    const float4* xr = (const float4*)(x + (size_t)row * Ln);
    float sum = 0.f;
    for (int i = threadIdx.x; i < Ln / 4; i += 256) {
        float4 v = xr[i];
        sum += (v.x + v.y) + (v.z + v.w);
    }
    // wave32 reduction
    for (int off = 16; off > 0; off >>= 1) sum += __shfl_down(sum, off, 32);
    __shared__ float red[8];
    if ((threadIdx.x & 31) == 0) red[threadIdx.x >> 5] = sum;
    __syncthreads();
    if (threadIdx.x == 0) {
        float t = 0.f;
        #pragma unroll
        for (int w = 0; w < 8; ++w) t += red[w];
        s_mean[row] = t * (1.0f / (float)Ln);
    }
}

// ---------------------------------------------------------------------------
// Kernel 2: MLP with V_WMMA_F32_16X16X4_F32 (confirmed lowering, wmma=12).
//   GEMM1: h[m,k]      = relu(sum_c A[m,c]*w1[c,k] + b1[k]),  A[m=4b+r, c] = s_mean[b,c]
//   GEMM2: scores[m,c] = sum_k h[m,k]*w2[k,c] + b2[c]
// Single 256-thread (8-wave) workgroup; h staged in LDS between GEMMs.
// ---------------------------------------------------------------------------
__global__ __launch_bounds__(256) void mlp_kernel(const float* __restrict__ s_mean,
                                                  const float* __restrict__ w1,
                                                  const float* __restrict__ b1,
                                                  const float* __restrict__ w2,
                                                  const float* __restrict__ b2,
                                                  float* __restrict__ scores) {
    __shared__ float hlds[Mrows * REDn];   // 64x16 f32 = 4 KB
    const int tid  = threadIdx.x;
    const int wave = tid >> 5;
    const int lane = tid & 31;
    const int half = lane >> 4;    // 0: K/M low half, 1: K+2 / M+8 half
    const int lm   = lane & 15;    // A-row (M) or B-col (N) within tile

    // ---- GEMM1: 4 M-tiles of 16 rows, waves 0..3, K=256 in steps of 4 ----
    if (wave < 4) {
        const int mrow = wave * 16 + lm;   // global row = b*4 + r
        const int bidx = mrow >> 2;        // batch (rows are r-replicated)
        v8f acc = {};
        for (int k0 = 0; k0 < Cn; k0 += 4) {
            const int ka = k0 + 2 * half;
            v2f a, bf;
            a.x  = s_mean[bidx * Cn + ka];
            a.y  = s_mean[bidx * Cn + ka + 1];
            bf.x = w1[ka * REDn + lm];
            bf.y = w1[(ka + 1) * REDn + lm];
            acc = __builtin_amdgcn_wmma_f32_16x16x4_f32(
                      false, a, false, bf, (short)0, acc, false, false);
        }
        const float bias = b1[lm];
        #pragma unroll
        for (int v = 0; v < 8; ++v) {
            const int m = wave * 16 + v + 8 * half;
            const float hv = acc[v] + bias;
            hlds[m * REDn + lm] = hv > 0.f ? hv : 0.f;   // relu
        }
    }
    __syncthreads();

    // ---- GEMM2: 4 (M) x 16 (N) tiles = 64 tiles over 8 waves, K=16 ----
    for (int t = wave; t < 64; t += 8) {
        const int tm = t >> 4;   // M tile
        const int tn = t & 15;   // N tile
        v8f acc = {};
        #pragma unroll
        for (int k0 = 0; k0 < REDn; k0 += 4) {
            const int ka = k0 + 2 * half;
            v2f a, bf;
            a.x  = hlds[(tm * 16 + lm) * REDn + ka];
            a.y  = hlds[(tm * 16 + lm) * REDn + ka + 1];
            bf.x = w2[ka * Cn + tn * 16 + lm];
            bf.y = w2[(ka + 1) * Cn + tn * 16 + lm];
            acc = __builtin_amdgcn_wmma_f32_16x16x4_f32(
                      false, a, false, bf, (short)0, acc, false, false);
        }
        const float bias = b2[tn * 16 + lm];
        #pragma unroll
        for (int v = 0; v < 8; ++v) {
            const int m = tm * 16 + v + 8 * half;
            scores[m * Cn + tn * 16 + lm] = acc[v] + bias;
        }
    }
}

// ---------------------------------------------------------------------------
// Kernel 3: x_f[b,c,l] = sum_r softmax_r(scores)[b,c,r] * x[b,c,rot_idx[r,l]].
// Memory-bound pass. Row staging now goes through the Tensor Data Mover:
// one 1-D D# per workgroup copies CT*Ln contiguous f32 (64 KB) global->LDS,
// tracked with TENSORcnt, while the wave prefetches the idx tables.
// ---------------------------------------------------------------------------
constexpr int CT      = 4;              // rows per block
constexpr int TILE_EL = CT * Ln;        // 16384 f32 = 64 KB (tile_dim0 fits 16 bits)

__device__ __forceinline__ void tdm_load_rows_to_lds(const float* gsrc, void* lds_dst) {
    const unsigned long long ga = (unsigned long long)(size_t)gsrc;
    const u32 lds_off = (u32)(size_t)lds_dst;     // addr[31:0] == LDS byte offset

    // D# group 0: count=1 (valid), lds_addr, global_addr[56:0], type=2
    u32x4 g0;
    g0.x = 1u;                                    // count=1, is_restore=0
    g0.y = lds_off;                               // lds_addr (bytes)
    g0.z = (u32)(ga & 0xFFFFFFFFull);             // global_addr[31:0]
    g0.w = (u32)((ga >> 32) & 0x01FFFFFFull)      // global_addr[56:32]
         | (2u << 30);                            // type=2 ("image")

    // D# group 1: data_size=4B, 1-D tensor/tile of TILE_EL elements
    i32x8 g1;
    g1[0] = (int)(2u << 16);                      // workgroup_mask=0, data_size=2 (4B)
    g1[1] = (int)(((u32)TILE_EL & 0xFFFFu) << 16);        // tensor_dim0[15:0]
    g1[2] = (int)(((u32)TILE_EL >> 16) | (1u << 16));     // tensor_dim0[31:16], tensor_dim1=1
    g1[3] = (int)(((u32)TILE_EL & 0xFFFFu) << 16);        // tensor_dim1 hi=0, tile_dim0
    g1[4] = 0;                                    // tile_dim1=0 (unused), tile_dim2=0
    g1[5] = (int)(u32)TILE_EL;                    // tensor_dim0_stride[31:0]
    g1[6] = 0;                                    // stride hi, tensor_dim1_stride lo
    g1[7] = 0;

    i32x4 gz = {0, 0, 0, 0};                      // groups 2/3 unused (<=2D tensor)

#if __clang_major__ >= 23
    i32x8 gz8 = {0, 0, 0, 0, 0, 0, 0, 0};
    __builtin_amdgcn_tensor_load_to_lds(g0, g1, gz, gz, gz8, 0);
#else
    __builtin_amdgcn_tensor_load_to_lds(g0, g1, gz, gz, 0);
#endif
}

__global__ __launch_bounds__(256) void combine_kernel(const float* __restrict__ x,
                                                      const int*   __restrict__ rot_idx,
                                                      const float* __restrict__ scores,
                                                      float*       __restrict__ out) {
    __shared__ float rows[TILE_EL];          // 64 KB
    const int bpc = Cn / CT;                 // 64 blocks per batch
    const int b   = blockIdx.x / bpc;
    const int c0  = (blockIdx.x % bpc) * CT;

    // TDM-stage CT contiguous rows; one DMA per workgroup (wave 0 issues it).
    if (threadIdx.x < 32) {
        tdm_load_rows_to_lds(x + ((size_t)b * Cn + c0) * Ln, (void*)rows);
    }
    // Pull the idx tables toward the caches while the DMA runs.
    __builtin_prefetch(rot_idx + threadIdx.x * 16, 0, 1);
    if (threadIdx.x < 32) {
        __builtin_amdgcn_s_wait_tensorcnt(0);
    }
    __syncthreads();

    for (int ci = 0; ci < CT; ++ci) {
        const int c = c0 + ci;
        // softmax over the 4 rotation scores (bitwise-equal -> exactly 0.25,
        // but computed faithfully from the WMMA results)
        float sc0 = scores[(b * Rn + 0) * Cn + c];
        float sc1 = scores[(b * Rn + 1) * Cn + c];
        float sc2 = scores[(b * Rn + 2) * Cn + c];
        float sc3 = scores[(b * Rn + 3) * Cn + c];
        float mx  = fmaxf(fmaxf(sc0, sc1), fmaxf(sc2, sc3));
        float e0 = __expf(sc0 - mx), e1 = __expf(sc1 - mx);
        float e2 = __expf(sc2 - mx), e3 = __expf(sc3 - mx);
        float inv = 1.0f / (e0 + e1 + e2 + e3);
        const float wr0 = e0 * inv, wr1 = e1 * inv, wr2 = e2 * inv, wr3 = e3 * inv;

        const float* rowp = rows + ci * Ln;
        float* op = out + ((size_t)b * Cn + c) * Ln;
        for (int l = threadIdx.x; l < Ln; l += 256) {
            const int i0 = rot_idx[0 * Ln + l];
            const int i1 = rot_idx[1 * Ln + l];
            const int i2 = rot_idx[2 * Ln + l];
            const int i3 = rot_idx[3 * Ln + l];
            op[l] = wr0 * rowp[i0] + wr1 * rowp[i1]
                  + wr2 * rowp[i2] + wr3 * rowp[i3];
        }
    }
}

// ---------------------------------------------------------------------------
extern "C" void kernel_launch(void* const* d_in, const int* in_sizes, int n_in,
                              void* d_out, int out_size, void* d_ws, size_t ws_size,
                              hipStream_t stream) {
    const float* x       = (const float*)d_in[0];
    const int*   rot_idx = (const int*)  d_in[1];
    const float* w1      = (const float*)d_in[2];
    const float* b1      = (const float*)d_in[3];
    const float* w2      = (const float*)d_in[4];
    const float* b2      = (const float*)d_in[5];
    float* out = (float*)d_out;

    float* s_mean = (float*)d_ws;                       // Bn*Cn    = 4096 f32
    float* scores = s_mean + Bn * Cn;                   // Mrows*Cn = 16384 f32

    mean_kernel<<<Bn * Cn, 256, 0, stream>>>(x, s_mean);
    mlp_kernel<<<1, 256, 0, stream>>>(s_mean, w1, b1, w2, b2, scores);
    combine_kernel<<<Bn * (Cn / CT), 256, 0, stream>>>(x, rot_idx, scores, out);
}